// UserInteractionPredictor_29506425323888
// MI455X (gfx1250) — compile-verified
//
#include <hip/hip_runtime.h>
#include <hip/hip_bf16.h>

typedef float v2f __attribute__((ext_vector_type(2)));
typedef float v8f __attribute__((ext_vector_type(8)));

#define DIM 128
#define N_PAIRS 524288
#define TILES_PER_SET (N_PAIRS / 16) /* 32768 tiles of 16 pairs */
#define WAVES_PER_BLOCK 8
#define EPSV 1e-8f

// sum value held on lanes L and L^16 (ds_swizzle SWAPX16: xor=0x10,and=0x1f)
__device__ __forceinline__ float sum_xor16(float x) {
    int yi = __builtin_amdgcn_ds_swizzle(__builtin_bit_cast(int, x), 0x401f);
    return x + __builtin_bit_cast(float, yi);
}

__global__ __launch_bounds__(256)
void UserInteractionPredictor_scores(const float* __restrict__ user_tab,
                                     const float* __restrict__ event_tab,
                                     const int*   __restrict__ pos_uidx,
                                     const int*   __restrict__ pos_eidx,
                                     const int*   __restrict__ neg_uidx,
                                     const int*   __restrict__ neg_eidx,
                                     const float* __restrict__ W,
                                     const float* __restrict__ b,
                                     float*       __restrict__ out)
{
    const int lane = threadIdx.x & 31;
    const int wave = threadIdx.x >> 5;
    const int tile = blockIdx.x * WAVES_PER_BLOCK + wave;  // one 16-pair tile per wave

    const int n = lane & 15;   // pair-in-tile (also WMMA column / A row)
    const int g = lane >> 4;   // K-half select within 4-wide chunk

    const bool is_pos = (tile < TILES_PER_SET);
    const int* uix = is_pos ? pos_uidx : neg_uidx;
    const int* eix = is_pos ? pos_eidx : neg_eidx;
    const int pair0 = (is_pos ? tile : tile - TILES_PER_SET) * 16;

    const size_t urow = (size_t)uix[pair0 + n] * DIM;
    const size_t erow = (size_t)eix[pair0 + n] * DIM;
    const int off = 2 * g;

    v8f acc = {};                          // U * E^T ; diagonal = dot(u_i, e_i)
    float suu = 0.f, see = 0.f, suw = 0.f; // per-pair partials (split over lane, lane^16)

    #pragma unroll 8
    for (int k = 0; k < DIM / 4; ++k) {
        const int o = 4 * k + off;
        v2f u2 = *(const v2f*)(user_tab  + urow + o);  // A layout: lane L -> u[L%16][4k+2g+{0,1}]
        v2f e2 = *(const v2f*)(event_tab + erow + o);  // B layout: lane L -> e[L%16][4k+2g+{0,1}]
        v2f w2 = *(const v2f*)(W + o);                 // tiny, L1/WGP$-resident

        acc = __builtin_amdgcn_wmma_f32_16x16x4_f32(false, u2, false, e2,
                                                    (short)0, acc, false, false);
        suu = fmaf(u2.x, u2.x, fmaf(u2.y, u2.y, suu));
        see = fmaf(e2.x, e2.x, fmaf(e2.y, e2.y, see));
        suw = fmaf(u2.x, w2.x, fmaf(u2.y, w2.y, suw));
    }

    // lanes L and L^16 jointly cover pair n's full 128 elements
    suu = sum_xor16(suu);
    see = sum_xor16(see);
    suw = sum_xor16(suw);

    // diag of f32 16x16 C/D: row M = v + 8*g, col N = n  ->  valid when
    // (g==0 && n<8, VGPR n) or (g==1 && n>=8, VGPR n-8)
    const int vsel = n - 8 * g;
    float sue = 0.f;
    #pragma unroll
    for (int j = 0; j < 8; ++j)
        if (vsel == j) sue = acc[j];

    const bool has_diag = (g == 0) ? (n < 8) : (n >= 8);
    if (has_diag) {
        float un  = fmaxf(sqrtf(suu), EPSV);
        float en  = fmaxf(sqrtf(see), EPSV);
        float sim = sue / (un * en);
        float score = fmaf(sim, W[DIM], suw + b[0]);
        out[(size_t)tile * 16 + n] = score;   // [pos scores | neg scores]
    }
}

extern "C" void kernel_launch(void* const* d_in, const int* in_sizes, int n_in,
                              void* d_out, int out_size, void* d_ws, size_t ws_size,
                              hipStream_t stream) {
    const float* user_tab  = (const float*)d_in[0];
    const float* event_tab = (const float*)d_in[1];
    const int*   pos_uidx  = (const int*)d_in[2];
    const int*   pos_eidx  = (const int*)d_in[3];
    const int*   neg_uidx  = (const int*)d_in[4];
    const int*   neg_eidx  = (const int*)d_in[5];
    const float* W         = (const float*)d_in[6];
    const float* b         = (const float*)d_in[7];
    float*       out       = (float*)d_out;

    const int total_tiles = 2 * TILES_PER_SET;                // 65536 waves
    dim3 grid(total_tiles / WAVES_PER_BLOCK);                 // 8192 blocks
    dim3 block(WAVES_PER_BLOCK * 32);                         // 256 threads (8 wave32)
    UserInteractionPredictor_scores<<<grid, block, 0, stream>>>(
        user_tab, event_tab, pos_uidx, pos_eidx, neg_uidx, neg_eidx, W, b, out);
}